// LeakyRNN_19516331393267
// MI455X (gfx1250) — compile-verified
//
#include <hip/hip_runtime.h>
#include <math.h>

typedef float v2f __attribute__((ext_vector_type(2)));
typedef float v8f __attribute__((ext_vector_type(8)));

#define T_LEN   64
#define B_SIZE  32
#define I_SIZE  128
#define H_SIZE  256
#define O_SIZE  32
#define NGRP    8
#define ALPHA_X 0.2f
#define OMX     0.8f
#define OMW     0.98f
#define EH      204      // sign boundary: +1 below, -1 at/above
#define NWG_H   16
#define NWG_B   2
#define NWG     (NWG_H*NWG_B)
#define BLK     256
#define IPAD    (I_SIZE+4)   // row stride stays 16B aligned
#define HPAD    (H_SIZE+4)

__device__ __forceinline__ float colmask(int j) {
  // mask_o / mask_a columns: +1 (j<204), -1 (j==204), 0 (j>=205)
  return (j < EH) ? 1.0f : ((j == EH) ? -1.0f : 0.0f);
}
__device__ __forceinline__ float hsign(int j) { return (j < EH) ? 1.0f : -1.0f; }

__global__ void zero_cnt_kernel(unsigned* cnt) {
  if (threadIdx.x == 0)
    __hip_atomic_store(cnt, 0u, __ATOMIC_RELEASE, __HIP_MEMORY_SCOPE_AGENT);
}

__device__ __forceinline__ void global_barrier(unsigned* cnt, unsigned target) {
  __threadfence();
  __syncthreads();
  if (threadIdx.x == 0) {
    __hip_atomic_fetch_add(cnt, 1u, __ATOMIC_ACQ_REL, __HIP_MEMORY_SCOPE_AGENT);
    while (__hip_atomic_load(cnt, __ATOMIC_ACQUIRE, __HIP_MEMORY_SCOPE_AGENT) < target) {
      __builtin_amdgcn_s_sleep(2);
    }
  }
  __syncthreads();
  __threadfence();
}

__global__ __launch_bounds__(BLK, 1) void leaky_rnn_persistent(
    const float* __restrict__ x, const float* __restrict__ Rs,
    const float* __restrict__ Wx, const float* __restrict__ Wh,
    const float* __restrict__ bh, const float* __restrict__ Wo,
    const float* __restrict__ Wa, const float* __restrict__ ba,
    const float* __restrict__ kappa,
    float* __restrict__ os_out, float* __restrict__ hs_out,
    float* __restrict__ wxs, float* __restrict__ whs, float* __restrict__ wos,
    unsigned* __restrict__ cnt)
{
  const int hi   = blockIdx.x;      // h-tile 0..15
  const int bhf  = blockIdx.y;      // b-half 0..1
  const int tid  = threadIdx.x;
  const int lane = tid & 31;
  const int wv   = tid >> 5;        // 8 waves
  const int bl   = tid >> 4;        // local batch 0..15  (phase C/D row owner)
  const int hl   = tid & 15;        // local h-row 0..15
  const int h0   = hi * 16;
  const int b0   = bhf * 16;
  const int bg   = b0 + bl;
  const int hg   = h0 + hl;

  __shared__ float s_wxpos[16][IPAD];   // relu(Wx) tile (WMMA A + clamp floor)
  __shared__ float s_whm [16][HPAD];    // relu(Wh)*mask_h tile (WMMA A; |.| = clamp floor)
  __shared__ float s_xe  [16][IPAD];    // xe_t        [b][i]
  __shared__ float s_xep [16][IPAD];    // xe_{t-1}    [b][i]
  __shared__ float s_outp[16][HPAD];    // out_{t-1}   [b][j]
  __shared__ float s_ti  [16][16];      // shared-GEMM tile [h][b]
  __shared__ float s_attn[16][NGRP];
  __shared__ float s_state[16][16];     // [b][h]
  __shared__ float s_noutp[16][16];     // new_out_{t-1} tile [b][h]
  __shared__ float s_valp[16][2];       // value_{t-1} [b][o-local]
  __shared__ float s_red [32][8];

  for (int q = tid; q < 16*I_SIZE; q += BLK) {
    int m = q >> 7, i = q & (I_SIZE-1);
    s_wxpos[m][i] = fmaxf(Wx[(h0+m)*I_SIZE + i], 0.f);
  }
  for (int q = tid; q < 16*H_SIZE; q += BLK) {
    int m = q >> 8, j = q & (H_SIZE-1);
    float wp = fmaxf(Wh[(h0+m)*H_SIZE + j], 0.f);
    s_whm[m][j] = wp * (((h0+m) == j) ? 0.f : hsign(j));
  }
  s_state[bl][hl] = 0.f;
  s_noutp[bl][hl] = 0.f;
  if (tid < 32) s_valp[tid >> 1][tid & 1] = 0.f;
  __syncthreads();

  const float kk0 = expf(kappa[0]);
  const float kk1 = expf(kappa[1]);
  const float kk2 = expf(kappa[2]);

  unsigned gen = 0;

  for (int t = 0; t < T_LEN; ++t) {
    ++gen;
    global_barrier(cnt, gen * NWG);   // hs[t-1], os[t-1], ws state visible

    // ---------------- Phase A: xep<-xe, stage out_prev, attention, xe ----
    for (int q = tid; q < 16*IPAD; q += BLK)
      (&s_xep[0][0])[q] = (t == 0) ? 0.f : (&s_xe[0][0])[q];
    __syncthreads();
    for (int q = tid; q < 16*H_SIZE; q += BLK) {
      int b = q >> 8, j = q & (H_SIZE-1);
      s_outp[b][j] = (t == 0) ? 0.f
                   : hs_out[((size_t)(t-1)*B_SIZE + (b0+b))*H_SIZE + j];
    }
    __syncthreads();
    if (tid < 16*NGRP) {                       // logits = out @ Wa_eff.T + ba
      int b = tid >> 3, g = tid & 7;
      float acc = ba[g];
      for (int h = 0; h < H_SIZE; ++h)
        acc += s_outp[b][h] * fmaxf(Wa[g*H_SIZE + h], 0.f) * colmask(h);
      s_attn[b][g] = acc;
    }
    __syncthreads();
    if (tid < 16) {                            // softmax over 8 groups
      float mx = s_attn[tid][0];
      for (int g = 1; g < NGRP; ++g) mx = fmaxf(mx, s_attn[tid][g]);
      float e[NGRP], sum = 0.f;
      for (int g = 0; g < NGRP; ++g) { e[g] = expf(s_attn[tid][g] - mx); sum += e[g]; }
      float inv = 1.f / sum;
      for (int g = 0; g < NGRP; ++g) s_attn[tid][g] = e[g] * inv;
    }
    __syncthreads();
    for (int q = tid; q < 16*I_SIZE; q += BLK) {  // xe = relu(x * attn_rep)
      int b = q >> 7, i = q & (I_SIZE-1);
      float xv = x[((size_t)t*B_SIZE + (b0+b))*I_SIZE + i];
      s_xe[b][i] = fmaxf(xv * s_attn[b][i >> 4], 0.f);
    }
    s_ti[tid & 15][tid >> 4] = 0.f;
    __syncthreads();

    // ---------------- Phase B: shared-weight GEMM via V_WMMA_F32_16X16X4 ----
    // ti[16h x 16b] = relu(Wx)tile @ xe  +  (relu(Wh)*mask)tile @ out_prev
    // K = 128 + 256 = 384, split 8 waves x 48 (12 wmma k-steps each).
    // Branch-free: select the LDS base address (v_cndmask), not the code path.
    {
      v8f c = {0.f,0.f,0.f,0.f,0.f,0.f,0.f,0.f};
      const int m     = lane & 15;            // A row / B col (n==m numerically)
      const int khalf = (lane >> 4) << 1;     // lanes 16-31 hold K+2 (ISA 7.12.2)
      const int kbase = wv * 48;
      const float* aWx = (const float*)s_wxpos + m*IPAD;
      const float* aWh = (const float*)s_whm  + m*HPAD - I_SIZE;  // aWh[ka]=s_whm[m][ka-128]
      const float* bXe = (const float*)s_xe   + m*IPAD;
      const float* bOu = (const float*)s_outp + m*HPAD - I_SIZE;
#pragma unroll
      for (int s = 0; s < 12; ++s) {
        const int ka = kbase + s*4 + khalf;   // ka<128 uniform per wave (4 | 128)
        const float* ap = (ka < I_SIZE) ? aWx : aWh;
        const float* bp = (ka < I_SIZE) ? bXe : bOu;
        v2f a  = *(const v2f*)(ap + ka);
        v2f b2 = *(const v2f*)(bp + ka);
        c = __builtin_amdgcn_wmma_f32_16x16x4_f32(false, a, false, b2,
                                                  (short)0, c, false, false);
      }
      const int n  = lane & 15;
      const int mo = (lane >> 4) * 8;         // C/D: lanes 16-31 hold M+8
#pragma unroll
      for (int r = 0; r < 8; ++r)
        atomicAdd(&s_ti[mo + r][n], c[r]);    // cross-wave K reduction (ds_add_f32)
    }
    __syncthreads();

    // ---------------- Phase C: fused plastic update(t-1) + matvec(t) ----
    // thread (bl,hl) owns rows wx[bg,hg,:] and wh[bg,hg,:]
    float accP = 0.f;
    const float Rprev  = (t >= 1) ? Rs[(t-1)*B_SIZE + bg] : 0.f;
    const float noutP  = s_noutp[bl][hl];
    const bool doUpd   = (t >= 1);
    const bool doRead  = (t >= 2);
    const bool doWrite = (t >= 1) && (t < T_LEN-1);
    {
      float* wrow = wxs + ((size_t)bg*H_SIZE + hg)*I_SIZE;
      const float cu = kk0 * Rprev * noutP;
      for (int i = 0; i < I_SIZE; i += 4) {
        float4 w = doRead ? *(const float4*)(wrow + i) : make_float4(0.f,0.f,0.f,0.f);
        if (doUpd) {
          w.x = fmaxf(w.x*OMW + cu*s_xep[bl][i+0], -s_wxpos[hl][i+0]);
          w.y = fmaxf(w.y*OMW + cu*s_xep[bl][i+1], -s_wxpos[hl][i+1]);
          w.z = fmaxf(w.z*OMW + cu*s_xep[bl][i+2], -s_wxpos[hl][i+2]);
          w.w = fmaxf(w.w*OMW + cu*s_xep[bl][i+3], -s_wxpos[hl][i+3]);
        }
        accP += w.x*s_xe[bl][i+0] + w.y*s_xe[bl][i+1]
              + w.z*s_xe[bl][i+2] + w.w*s_xe[bl][i+3];
        if (doWrite) *(float4*)(wrow + i) = w;
        __builtin_prefetch((const void*)(wrow + i + 32), 0, 1);
      }
    }
    {
      float* wrow = whs + ((size_t)bg*H_SIZE + hg)*H_SIZE;
      const float cu = kk1 * Rprev * noutP;
      const float* o2row = hs_out + ((size_t)(t >= 2 ? t-2 : 0)*B_SIZE + bg)*H_SIZE;
      for (int j = 0; j < H_SIZE; j += 4) {
        float4 w, o2;
        if (doRead) { w = *(const float4*)(wrow + j); o2 = *(const float4*)(o2row + j); }
        else        { w = make_float4(0.f,0.f,0.f,0.f); o2 = make_float4(0.f,0.f,0.f,0.f); }
        if (doUpd) {
          // clamp floor -relu(Wh) == -|relu(Wh)*mask| off-diagonal; the diagonal
          // element is masked to zero in every output path, so its floor is
          // unobservable -> reuse the masked tile and save 16KB of LDS.
          w.x = fmaxf(w.x*OMW + cu*o2.x, -fabsf(s_whm[hl][j+0]));
          w.y = fmaxf(w.y*OMW + cu*o2.y, -fabsf(s_whm[hl][j+1]));
          w.z = fmaxf(w.z*OMW + cu*o2.z, -fabsf(s_whm[hl][j+2]));
          w.w = fmaxf(w.w*OMW + cu*o2.w, -fabsf(s_whm[hl][j+3]));
        }
        const float sg = hsign(j);            // 204 % 4 == 0: uniform per chunk
        float4 ov = *(const float4*)&s_outp[bl][j];
        accP += w.x*((hg==j+0)?0.f:sg)*ov.x + w.y*((hg==j+1)?0.f:sg)*ov.y
              + w.z*((hg==j+2)?0.f:sg)*ov.z + w.w*((hg==j+3)?0.f:sg)*ov.w;
        if (doWrite) *(float4*)(wrow + j) = w;
        __builtin_prefetch((const void*)(wrow + j + 32), 0, 1);
      }
    }

    // ---------------- Phase D: leaky state + tanh(relu) ----
    {
      const float total = s_ti[hl][bl] + accP + bh[hg];
      const float ns = s_state[bl][hl]*OMX + total*ALPHA_X;
      s_state[bl][hl] = ns;
      const float no = tanhf(fmaxf(ns, 0.f));
      hs_out[((size_t)t*B_SIZE + bg)*H_SIZE + hg] = no;
      s_noutp[bl][hl] = no;   // only ever read by this same thread
    }

    ++gen;
    global_barrier(cnt, gen * NWG);   // full new_out (hs[t]) visible everywhere

    // ---------------- Phase E: output head (2 o-rows per WG, fused wo) ----
    {
      const int row = tid >> 3, sub = tid & 7;
      const int ebl = row >> 1, eol = row & 1;
      const int ebg = b0 + ebl;
      const int o   = hi*2 + eol;
      float* wrow = wos + ((size_t)ebg*O_SIZE + o)*H_SIZE;
      const float vprev = s_valp[ebl][eol];
      const float Rp = (t >= 1) ? Rs[(t-1)*B_SIZE + ebg] : 0.f;
      const float cu = kk2 * ((Rp + 1.f)*0.5f - vprev);
      const float* n1row = hs_out + ((size_t)(t >= 1 ? t-1 : 0)*B_SIZE + ebg)*H_SIZE;
      const float* n0row = hs_out + ((size_t)t*B_SIZE + ebg)*H_SIZE;
      float acc = 0.f;
      const int jb = sub * 32;
      for (int j = jb; j < jb + 32; j += 4) {
        float4 w = doRead ? *(const float4*)(wrow + j) : make_float4(0.f,0.f,0.f,0.f);
        if (doUpd) {                 // wo has no clamp
          float4 n1 = *(const float4*)(n1row + j);
          w.x = w.x*OMW + cu*n1.x;  w.y = w.y*OMW + cu*n1.y;
          w.z = w.z*OMW + cu*n1.z;  w.w = w.w*OMW + cu*n1.w;
        }
        float4 wp;
        wp.x = fmaxf(Wo[o*H_SIZE + j+0], 0.f);
        wp.y = fmaxf(Wo[o*H_SIZE + j+1], 0.f);
        wp.z = fmaxf(Wo[o*H_SIZE + j+2], 0.f);
        wp.w = fmaxf(Wo[o*H_SIZE + j+3], 0.f);
        float4 n0 = *(const float4*)(n0row + j);
        acc += (wp.x + w.x)*colmask(j+0)*n0.x + (wp.y + w.y)*colmask(j+1)*n0.y
             + (wp.z + w.z)*colmask(j+2)*n0.z + (wp.w + w.w)*colmask(j+3)*n0.w;
        if (doWrite) *(float4*)(wrow + j) = w;
      }
      s_red[row][sub] = acc;
    }
    __syncthreads();
    if (tid < 32) {
      float s = 0.f;
#pragma unroll
      for (int q = 0; q < 8; ++q) s += s_red[tid][q];
      const int ebl = tid >> 1, eol = tid & 1;
      const float v = tanhf(fmaxf(s, 0.f));
      os_out[((size_t)t*B_SIZE + (b0+ebl))*O_SIZE + (hi*2 + eol)] = v;
      s_valp[ebl][eol] = v;
    }
    // next-iteration barrier orders s_red/s_valp reuse
  }
}

extern "C" void kernel_launch(void* const* d_in, const int* in_sizes, int n_in,
                              void* d_out, int out_size, void* d_ws, size_t ws_size,
                              hipStream_t stream) {
  const float* x  = (const float*)d_in[0];
  const float* Rs = (const float*)d_in[1];
  const float* Wx = (const float*)d_in[2];
  const float* Wh = (const float*)d_in[3];
  const float* bh = (const float*)d_in[4];
  const float* Wo = (const float*)d_in[5];
  const float* Wa = (const float*)d_in[6];
  const float* ba = (const float*)d_in[7];
  const float* kp = (const float*)d_in[8];

  float* osb = (float*)d_out;                       // (T,B,O) = 65536 floats
  float* hsb = osb + (size_t)T_LEN*B_SIZE*O_SIZE;   // (T,B,H)

  // ws layout (floats): wx 1,048,576 | wh 2,097,152 | wo 262,144 | barrier cnt
  float* wsf = (float*)d_ws;
  float*    wxs = wsf;
  float*    whs = wsf + 1048576;
  float*    wos = wsf + 3145728;
  unsigned* cnt = (unsigned*)(wsf + 3407872);

  zero_cnt_kernel<<<1, 32, 0, stream>>>(cnt);
  leaky_rnn_persistent<<<dim3(NWG_H, NWG_B), BLK, 0, stream>>>(
      x, Rs, Wx, Wh, bh, Wo, Wa, ba, kp, osb, hsb, wxs, whs, wos, cnt);
}